// LNO2D_9234179687354
// MI455X (gfx1250) — compile-verified
//
#include <hip/hip_runtime.h>

// AFNO-2D mixer for MI455X (gfx1250).
// fp32 WMMA (v_wmma_f32_16x16x4_f32) for all GEMMs, LDS period-128 twiddle table,
// Tensor Data Mover (tensor_load_to_lds / tensor_store_from_lds, TENSORcnt) for
// all 2-D slab staging, transposed LDS layouts to keep fragment reads conflict-free.
// Scratch requirement: ws_size >= 2 * 4*128*65*768 * 4 bytes = 204,472,320 bytes.

#define Bn 4
#define Hn 128
#define Wn 128
#define Cn 768
#define WM 65            // Wn/2 + 1
#define NH 8
#define BL 96            // Cn / NH
#define SPEC ((size_t)Bn * Hn * WM * Cn)   // floats per spectral plane

#define TWO_PI_OVER_128 0.049087385212340517f
#define LAMBDA 0.01f

typedef __attribute__((ext_vector_type(2))) float     v2f;
typedef __attribute__((ext_vector_type(8))) float     v8f;
typedef __attribute__((ext_vector_type(4))) unsigned  v4u;
typedef __attribute__((ext_vector_type(4))) int       v4i;
typedef __attribute__((ext_vector_type(8))) int       v8i;

#ifndef __has_builtin
#define __has_builtin(x) 0
#endif
#if __has_builtin(__builtin_amdgcn_tensor_load_to_lds) && \
    __has_builtin(__builtin_amdgcn_tensor_store_from_lds) && \
    __has_builtin(__builtin_amdgcn_s_wait_tensorcnt)
#define HAVE_TDM 1
#else
#define HAVE_TDM 0
#endif

// D = A(16x4) * B(4x16) + C, fp32 WMMA. NEG[1:0] must be 0 for float types (ISA 7.12);
// signs are folded into the A-fragment values (twiddle table stores e^{-i*theta}).
__device__ __forceinline__ v8f wmma4(v2f a, v2f b, v8f c) {
  return __builtin_amdgcn_wmma_f32_16x16x4_f32(false, a, false, b, (short)0, c, false, false);
}

__device__ __forceinline__ float softshrink(float v) {
  return v > LAMBDA ? v - LAMBDA : (v < -LAMBDA ? v + LAMBDA : 0.0f);
}

// Low 32 bits of a flat shared-aperture address ARE the LDS byte offset (ISA 10.2).
__device__ __forceinline__ unsigned lds_lo(const void* p) {
  return (unsigned)(unsigned long long)(size_t)p;
}

#if HAVE_TDM
// Tensor DMA descriptor per ISA ch.8: 2-D tile, 4-byte elements, no cluster multicast,
// no padding/iteration/gather. tile/tensor dims + stride in elements.
__device__ __forceinline__ void tdm_xfer(bool store, unsigned lds_addr,
                                         unsigned long long ga,
                                         unsigned tile0, unsigned tile1,
                                         unsigned ten0, unsigned ten1,
                                         unsigned long long stride0) {
  v4u g0;
  g0.x = 1u;                                             // count=1, user mode
  g0.y = lds_addr;                                       // lds_addr[31:0]
  g0.z = (unsigned)(ga & 0xffffffffull);                 // global_addr[31:0]
  g0.w = (unsigned)((ga >> 32) & 0x1ffffffull) | (2u << 30);  // addr[56:32] | type=2
  v8i g1;
  g1[0] = 0x00020000;                                    // data_size=2 (4 bytes)
  g1[1] = (int)((ten0 & 0xffffu) << 16);                 // tensor_dim0[15:0]
  g1[2] = (int)((ten0 >> 16) | ((ten1 & 0xffffu) << 16));// tensor_dim0[31:16] | dim1[15:0]
  g1[3] = (int)((ten1 >> 16) | (tile0 << 16));           // tensor_dim1[31:16] | tile_dim0
  g1[4] = (int)(tile1 & 0xffffu);                        // tile_dim1 (tile_dim2=0 -> 2D)
  g1[5] = (int)(stride0 & 0xffffffffull);                // tensor_dim0_stride[31:0]
  g1[6] = (int)((stride0 >> 32) & 0xffffull);            // stride[47:32]
  g1[7] = 0;
  v4i gz = {0, 0, 0, 0};
#if __clang_major__ >= 23
  v8i gz8 = {0, 0, 0, 0, 0, 0, 0, 0};
  if (store) __builtin_amdgcn_tensor_store_from_lds(g0, g1, gz, gz, gz8, 0);
  else       __builtin_amdgcn_tensor_load_to_lds(g0, g1, gz, gz, gz8, 0);
#else
  if (store) __builtin_amdgcn_tensor_store_from_lds(g0, g1, gz, gz, 0);
  else       __builtin_amdgcn_tensor_load_to_lds(g0, g1, gz, gz, 0);
#endif
}
#endif

// ---------------------------------------------------------------------------
// Kernel 1: row rFFT as GEMM.  S[k,c] = sum_n e^{-2*pi*i*n*k/128} x[n,c]
// TDM stages the (128 x 96 f32, row-stride 768) x-slab; twiddles from LDS table.
// ---------------------------------------------------------------------------
__global__ __launch_bounds__(256)
void afno_rowfft(const float* __restrict__ x,
                 float* __restrict__ Sr, float* __restrict__ Si) {
  __shared__ float  xs[128][96];
  __shared__ float2 tw[128];                 // tw[m] = e^{-2*pi*i*m/128}
  const int tid   = threadIdx.x;
  const int bh    = blockIdx.x;              // b*128 + h
  const int cb    = blockIdx.y * BL;
  const int wave  = tid >> 5, lane = tid & 31;
  const int laneN = lane & 15, laneH = lane >> 4;

#if HAVE_TDM
  if (wave == 0)
    tdm_xfer(false, lds_lo(&xs[0][0]),
             (unsigned long long)(size_t)(x + (size_t)bh * 128 * Cn + cb),
             96, 128, 96, 128, Cn);
#else
  for (int i = tid; i < 128 * BL; i += 256) {
    int n = i / BL, cc = i - n * BL;
    xs[n][cc] = x[(size_t)(bh * 128 + n) * Cn + cb + cc];
  }
#endif
  if (tid < 128) {
    float sn, cs;
    __sincosf(TWO_PI_OVER_128 * (float)tid, &sn, &cs);
    tw[tid] = make_float2(cs, -sn);
  }
#if HAVE_TDM
  if (wave == 0) __builtin_amdgcn_s_wait_tensorcnt(0);
#endif
  __syncthreads();

  // 5 k-tiles (k = 0..79, masked at 65) x 6 c-tiles
  for (int t = wave; t < 5 * 6; t += 8) {
    const int m0 = (t / 6) * 16, n0 = (t % 6) * 16;
    const int kout = m0 + laneN, ci = n0 + laneN;
    int idx = (2 * laneH * kout) & 127;          // running (n*k) mod 128
    v8f accR = {}, accI = {};
    for (int s = 0; s < 32; ++s) {
      const int nsp = 4 * s + 2 * laneH;
      float2 t0 = tw[idx];
      float2 t1 = tw[(idx + kout) & 127];
      idx = (idx + 4 * kout) & 127;
      v2f aR = {t0.x, t1.x};
      v2f aI = {t0.y, t1.y};
      v2f bv = {xs[nsp][ci], xs[nsp + 1][ci]};
      accR = wmma4(aR, bv, accR);
      accI = wmma4(aI, bv, accI);
    }
    for (int r = 0; r < 8; ++r) {
      const int k = m0 + r + 8 * laneH;
      if (k < WM) {
        const int g = (bh * WM + k) * Cn + cb + ci;   // < 2^31, saddr+voffset form
        Sr[g] = accR[r];
        Si[g] = accI[r];
      }
    }
  }
}

// ---------------------------------------------------------------------------
// Kernel 2: per (b, w-mode, head) column pipeline, fully in LDS:
//   TDM load slab -> col DFT -> complex MLP1(ReLU) -> MLP2 + softshrink
//   -> inverse col DFT (x 1/128) -> TDM store back in place.
// LDS regions (floats):
//   R0: natural [2][128][96]  (S slab in, Z slab out -> TDM source/target)
//   R1: Xt transposed [2][96][132]   (later reused as natural Y [2][128][96])
//   R2: o1t transposed [2][96][132]
//   TW: 128 x float2 twiddle table
// ---------------------------------------------------------------------------
#define R0PLANE (128 * 96)                 // 12288
#define R1OFF   (2 * R0PLANE)              // 24576
#define TP      132                        // transposed pitch: 4-bank lane stride
#define R1PLANE (96 * TP)                  // 12672
#define R2OFF   (R1OFF + 2 * R1PLANE)      // 49920
#define TWOFF   (R2OFF + 2 * R1PLANE)      // 75264
#define SMEM2   ((TWOFF + 256) * 4)        // 302080 bytes <= 320 KB/WGP

__global__ __launch_bounds__(256)
void afno_colmix(float* __restrict__ Sr, float* __restrict__ Si,
                 const float* __restrict__ w1, const float* __restrict__ b1,
                 const float* __restrict__ w2, const float* __restrict__ b2) {
  extern __shared__ float sm[];
  float*  S0  = sm;                        // natural re
  float*  S1  = sm + R0PLANE;              // natural im
  float*  Xt  = sm + R1OFF;                // transposed re (+R1PLANE = im)
  float*  O1t = sm + R2OFF;                // transposed re (+R1PLANE = im)
  float*  Y0  = sm + R1OFF;                // natural re (reuses R1 after Xt dies)
  float*  Y1  = Y0 + R0PLANE;              // natural im
  float2* tw  = (float2*)(sm + TWOFF);

  const int tid  = threadIdx.x;
  const int wk   = blockIdx.x;             // w-mode 0..64
  const int head = blockIdx.y;
  const int b    = blockIdx.z;
  const int rowStride = WM * Cn;           // 49920
  const int gbase = ((b * Hn) * WM + wk) * Cn + head * BL;
  const int wave  = tid >> 5, lane = tid & 31;
  const int laneN = lane & 15, laneH = lane >> 4;

  // --- stage 0: async-stage the complex column slab + build twiddle table ---
#if HAVE_TDM
  if (wave == 0) {
    tdm_xfer(false, lds_lo(S0), (unsigned long long)(size_t)(Sr + gbase),
             96, 128, 96, 128, rowStride);
    tdm_xfer(false, lds_lo(S1), (unsigned long long)(size_t)(Si + gbase),
             96, 128, 96, 128, rowStride);
  }
#else
  for (int i = tid; i < 128 * BL; i += 256) {
    int h = i / BL, cc = i - h * BL;
    int g = gbase + h * rowStride + cc;
    S0[h * 96 + cc] = Sr[g];
    S1[h * 96 + cc] = Si[g];
  }
#endif
  if (tid < 128) {
    float sn, cs;
    __sincosf(TWO_PI_OVER_128 * (float)tid, &sn, &cs);
    tw[tid] = make_float2(cs, -sn);
  }
#if HAVE_TDM
  if (wave == 0) __builtin_amdgcn_s_wait_tensorcnt(0);
#endif
  __syncthreads();

  // --- stage 1: forward column DFT, S(natural) -> Xt(transposed) ---
  for (int t = wave; t < 48; t += 8) {
    const int m0 = (t / 6) * 16, n0 = (t % 6) * 16;
    const int kout = m0 + laneN, ci = n0 + laneN;
    int idx = (2 * laneH * kout) & 127;
    v8f Prr = {}, Pii = {}, Pri = {}, Pir = {};
    for (int s = 0; s < 32; ++s) {
      const int h0 = 4 * s + 2 * laneH;
      float2 t0 = tw[idx];
      float2 t1 = tw[(idx + kout) & 127];
      idx = (idx + 4 * kout) & 127;
      v2f aR = {t0.x, t1.x}, aI = {t0.y, t1.y};
      v2f bR = {S0[h0 * 96 + ci], S0[(h0 + 1) * 96 + ci]};
      v2f bI = {S1[h0 * 96 + ci], S1[(h0 + 1) * 96 + ci]};
      Prr = wmma4(aR, bR, Prr);  Pii = wmma4(aI, bI, Pii);
      Pri = wmma4(aR, bI, Pri);  Pir = wmma4(aI, bR, Pir);
    }
    for (int r = 0; r < 8; ++r) {
      const int k = m0 + r + 8 * laneH;
      Xt[ci * TP + k]           = Prr[r] - Pii[r];   // Re, [c][k]
      Xt[R1PLANE + ci * TP + k] = Pri[r] + Pir[r];   // Im
    }
  }
  __syncthreads();

  // --- stage 2: MLP layer 1, o1 = relu(X*W1 + b1); Xt -> O1t ---
  {
    const float* w1r = w1 + head * BL * BL;          // w1[0][head]
    const float* w1i = w1 + (NH + head) * BL * BL;   // w1[1][head]
    const float* b1r = b1 + head * BL;
    const float* b1i = b1 + (NH + head) * BL;
    for (int t = wave; t < 48; t += 8) {
      const int m0 = (t / 6) * 16, n0 = (t % 6) * 16;
      const int row = m0 + laneN, co = n0 + laneN;
      v8f Prr = {}, Pii = {}, Pri = {}, Pir = {};
      for (int s = 0; s < 24; ++s) {
        const int k0 = 4 * s + 2 * laneH;            // cin
        v2f aR = {Xt[k0 * TP + row], Xt[(k0 + 1) * TP + row]};
        v2f aI = {Xt[R1PLANE + k0 * TP + row], Xt[R1PLANE + (k0 + 1) * TP + row]};
        v2f bR = {w1r[k0 * BL + co], w1r[(k0 + 1) * BL + co]};
        v2f bI = {w1i[k0 * BL + co], w1i[(k0 + 1) * BL + co]};
        Prr = wmma4(aR, bR, Prr);  Pii = wmma4(aI, bI, Pii);
        Pri = wmma4(aR, bI, Pri);  Pir = wmma4(aI, bR, Pir);
      }
      const float br = b1r[co], bi = b1i[co];
      for (int r = 0; r < 8; ++r) {
        const int k = m0 + r + 8 * laneH;
        float orv = Prr[r] - Pii[r] + br;  orv = orv > 0.f ? orv : 0.f;
        float oiv = Pri[r] + Pir[r] + bi;  oiv = oiv > 0.f ? oiv : 0.f;
        O1t[co * TP + k]           = orv;
        O1t[R1PLANE + co * TP + k] = oiv;
      }
    }
  }
  __syncthreads();

  // --- stage 3: MLP layer 2 + softshrink; O1t -> Y(natural, in R1) ---
  {
    const float* w2r = w2 + head * BL * BL;
    const float* w2i = w2 + (NH + head) * BL * BL;
    const float* b2r = b2 + head * BL;
    const float* b2i = b2 + (NH + head) * BL;
    for (int t = wave; t < 48; t += 8) {
      const int m0 = (t / 6) * 16, n0 = (t % 6) * 16;
      const int row = m0 + laneN, co = n0 + laneN;
      v8f Prr = {}, Pii = {}, Pri = {}, Pir = {};
      for (int s = 0; s < 24; ++s) {
        const int k0 = 4 * s + 2 * laneH;
        v2f aR = {O1t[k0 * TP + row], O1t[(k0 + 1) * TP + row]};
        v2f aI = {O1t[R1PLANE + k0 * TP + row], O1t[R1PLANE + (k0 + 1) * TP + row]};
        v2f bR = {w2r[k0 * BL + co], w2r[(k0 + 1) * BL + co]};
        v2f bI = {w2i[k0 * BL + co], w2i[(k0 + 1) * BL + co]};
        Prr = wmma4(aR, bR, Prr);  Pii = wmma4(aI, bI, Pii);
        Pri = wmma4(aR, bI, Pri);  Pir = wmma4(aI, bR, Pir);
      }
      const float br = b2r[co], bi = b2i[co];
      for (int r = 0; r < 8; ++r) {
        const int k = m0 + r + 8 * laneH;
        Y0[k * 96 + co] = softshrink(Prr[r] - Pii[r] + br);
        Y1[k * 96 + co] = softshrink(Pri[r] + Pir[r] + bi);
      }
    }
  }
  __syncthreads();

  // --- stage 4: inverse column DFT (x 1/128); Y(natural) -> Z in R0, TDM store ---
  for (int t = wave; t < 48; t += 8) {
    const int m0 = (t / 6) * 16, n0 = (t % 6) * 16;
    const int hout = m0 + laneN, ci = n0 + laneN;
    int idx = (-(2 * laneH * hout)) & 127;           // conj: negate index
    v8f Prr = {}, Pii = {}, Pri = {}, Pir = {};
    for (int s = 0; s < 32; ++s) {
      const int k0 = 4 * s + 2 * laneH;
      float2 t0 = tw[idx];
      float2 t1 = tw[(idx - hout) & 127];
      idx = (idx - 4 * hout) & 127;
      v2f aR = {t0.x, t1.x}, aI = {t0.y, t1.y};      // = {cos, +sin}
      v2f bR = {Y0[k0 * 96 + ci], Y0[(k0 + 1) * 96 + ci]};
      v2f bI = {Y1[k0 * 96 + ci], Y1[(k0 + 1) * 96 + ci]};
      Prr = wmma4(aR, bR, Prr);  Pii = wmma4(aI, bI, Pii);
      Pri = wmma4(aR, bI, Pri);  Pir = wmma4(aI, bR, Pir);
    }
    for (int r = 0; r < 8; ++r) {
      const int h = m0 + r + 8 * laneH;
      S0[h * 96 + ci] = (Prr[r] - Pii[r]) * (1.0f / 128.0f);
      S1[h * 96 + ci] = (Pri[r] + Pir[r]) * (1.0f / 128.0f);
    }
  }
  __syncthreads();
#if HAVE_TDM
  if (wave == 0) {
    tdm_xfer(true, lds_lo(S0), (unsigned long long)(size_t)(Sr + gbase),
             96, 128, 96, 128, rowStride);
    tdm_xfer(true, lds_lo(S1), (unsigned long long)(size_t)(Si + gbase),
             96, 128, 96, 128, rowStride);
    __builtin_amdgcn_s_wait_tensorcnt(0);
  }
#else
  for (int i = tid; i < 128 * BL; i += 256) {
    int h = i / BL, cc = i - h * BL;
    int g = gbase + h * rowStride + cc;
    Sr[g] = S0[h * 96 + cc];
    Si[g] = S1[h * 96 + cc];
  }
#endif
}

// ---------------------------------------------------------------------------
// Kernel 3: inverse row rFFT (Hermitian fold, x 1/128) + residual add.
// out[n,c] = x[n,c] + sum_{k=0..64} s_k/128 * (cos Zr - sin Zi), s_k in {1,2}.
// TDM zero-fills the K pad 65->68 via tensor OOB semantics (reads past dim -> 0).
// ---------------------------------------------------------------------------
__global__ __launch_bounds__(256)
void afno_rowifft_bias(const float* __restrict__ Sr, const float* __restrict__ Si,
                       const float* __restrict__ x, float* __restrict__ out) {
  __shared__ float  zs[2][68][96];
  __shared__ float2 tw[128];
  const int tid   = threadIdx.x;
  const int bh    = blockIdx.x;
  const int cb    = blockIdx.y * BL;
  const int wave  = tid >> 5, lane = tid & 31;
  const int laneN = lane & 15, laneH = lane >> 4;

#if HAVE_TDM
  if (wave == 0) {
    // tile 96 x 68, tensor_dim1 = 65 -> rows 65..67 read as zero
    tdm_xfer(false, lds_lo(&zs[0][0][0]),
             (unsigned long long)(size_t)(Sr + (size_t)bh * WM * Cn + cb),
             96, 68, 96, 65, Cn);
    tdm_xfer(false, lds_lo(&zs[1][0][0]),
             (unsigned long long)(size_t)(Si + (size_t)bh * WM * Cn + cb),
             96, 68, 96, 65, Cn);
  }
#else
  for (int i = tid; i < 68 * BL; i += 256) {
    int k = i / BL, cc = i - k * BL;
    float vr = 0.f, vi = 0.f;
    if (k < WM) {
      int g = (bh * WM + k) * Cn + cb + cc;
      vr = Sr[g];  vi = Si[g];
    }
    zs[0][k][cc] = vr;
    zs[1][k][cc] = vi;
  }
#endif
  if (tid < 128) {
    float sn, cs;
    __sincosf(TWO_PI_OVER_128 * (float)tid, &sn, &cs);
    tw[tid] = make_float2(cs, -sn);
  }
#if HAVE_TDM
  if (wave == 0) __builtin_amdgcn_s_wait_tensorcnt(0);
#endif
  __syncthreads();

  for (int t = wave; t < 48; t += 8) {
    const int m0 = (t / 6) * 16, n0 = (t % 6) * 16;
    const int nsp = m0 + laneN, ci = n0 + laneN;
    int idx = (2 * laneH * nsp) & 127;
    v8f acc = {};
    for (int s = 0; s < 17; ++s) {
      const int k0 = 4 * s + 2 * laneH;
      const int k1 = k0 + 1;
      float sc0 = (k0 == 0 || k0 == 64) ? (1.0f / 128.0f)
                                        : (k0 < WM ? (2.0f / 128.0f) : 0.0f);
      float sc1 = (k1 == 64) ? (1.0f / 128.0f)
                             : (k1 < WM ? (2.0f / 128.0f) : 0.0f);
      float2 t0 = tw[idx];
      float2 t1 = tw[(idx + nsp) & 127];
      idx = (idx + 4 * nsp) & 127;
      v2f aR = {sc0 * t0.x, sc1 * t1.x};            // sc * cos
      v2f aI = {sc0 * t0.y, sc1 * t1.y};            // sc * (-sin)
      v2f bR = {zs[0][k0][ci], zs[0][k1][ci]};
      v2f bI = {zs[1][k0][ci], zs[1][k1][ci]};
      acc = wmma4(aR, bR, acc);
      acc = wmma4(aI, bI, acc);
    }
    for (int r = 0; r < 8; ++r) {
      const int n = m0 + r + 8 * laneH;
      const int g = (bh * Wn + n) * Cn + cb + ci;
      out[g] = acc[r] + x[g];
    }
  }
}

// ---------------------------------------------------------------------------
extern "C" void kernel_launch(void* const* d_in, const int* in_sizes, int n_in,
                              void* d_out, int out_size, void* d_ws, size_t ws_size,
                              hipStream_t stream) {
  (void)in_sizes; (void)n_in; (void)out_size; (void)ws_size;
  const float* x  = (const float*)d_in[0];
  const float* w1 = (const float*)d_in[1];
  const float* b1 = (const float*)d_in[2];
  const float* w2 = (const float*)d_in[3];
  const float* b2 = (const float*)d_in[4];
  float* out = (float*)d_out;

  float* Sr = (float*)d_ws;           // spectral real plane
  float* Si = Sr + SPEC;              // spectral imag plane

  // 1) row rFFT
  afno_rowfft<<<dim3(Bn * Hn, Cn / BL), 256, 0, stream>>>(x, Sr, Si);

  // 2) column DFT + complex MLP + softshrink + inverse column DFT (in place)
  (void)hipFuncSetAttribute(reinterpret_cast<const void*>(&afno_colmix),
                            hipFuncAttributeMaxDynamicSharedMemorySize, (int)SMEM2);
  afno_colmix<<<dim3(WM, NH, Bn), 256, SMEM2, stream>>>(Sr, Si, w1, b1, w2, b2);

  // 3) inverse row rFFT + residual
  afno_rowifft_bias<<<dim3(Bn * Hn, Cn / BL), 256, 0, stream>>>(Sr, Si, x, out);
}